// KUPA_72567767433689
// MI455X (gfx1250) — compile-verified
//
#include <hip/hip_runtime.h>
#include <math.h>

typedef float v2f __attribute__((ext_vector_type(2)));
typedef float v8f __attribute__((ext_vector_type(8)));

#define DEV __device__ __forceinline__

// Extract diagonal of the 16x16 f32 C/D tile (8 VGPRs, wave32 layout):
//   (i,i) lives at (vgpr=i, lane=i) for i<8 and (vgpr=i-8, lane=i+16) for i>=8.
// Every lane materializes c[lane&7], then lane j pulls from source lane
// (j<8 ? j : j+16). Lanes 0..15 end up holding dot[0..15].
DEV float wmma_diag(v8f c, int lane) {
  int sel = lane & 7;
  float e = c[0];
  e = (sel == 1) ? c[1] : e;
  e = (sel == 2) ? c[2] : e;
  e = (sel == 3) ? c[3] : e;
  e = (sel == 4) ? c[4] : e;
  e = (sel == 5) ? c[5] : e;
  e = (sel == 6) ? c[6] : e;
  e = (sel == 7) ? c[7] : e;
  int j = lane & 15;
  int src = (j < 8) ? j : (j + 16);
  return __shfl(e, src, 32);
}

// ---------------------------------------------------------------------------
// Pass 0: zero the aggregation outputs, init segment max = -inf, sum = 0.
// ---------------------------------------------------------------------------
__global__ __launch_bounds__(256) void init_kernel(
    float* __restrict__ out_zero, long n_zero,
    float* __restrict__ smax_e, float* __restrict__ ssum_e, int n_ent,
    float* __restrict__ smax_u, float* __restrict__ ssum_u, int n_usr) {
  long i = (long)blockIdx.x * blockDim.x + threadIdx.x;
  long stride = (long)gridDim.x * blockDim.x;
  const float NEG_INF = __int_as_float(0xff800000);
  for (long k = i; k < n_zero; k += stride) out_zero[k] = 0.0f;
  for (long k = i; k < n_ent; k += stride) { smax_e[k] = NEG_INF; ssum_e[k] = 0.0f; }
  for (long k = i; k < n_usr; k += stride) { smax_u[k] = NEG_INF; ssum_u[k] = 0.0f; }
}

// ---------------------------------------------------------------------------
// Pass 1 (KG): one wave computes 16 edge dot-products via 16 chained
// V_WMMA_F32_16X16X4_F32 (K=64 in steps of 4), true fp32 accumulation.
// Writes raw score and atomic-max per head segment (global_atomic_max_num_f32).
// ---------------------------------------------------------------------------
__global__ __launch_bounds__(256) void kg_score_kernel(
    const float* __restrict__ ent, const float* __restrict__ rel,
    const int* __restrict__ head, const int* __restrict__ tail,
    const int* __restrict__ etype,
    float* __restrict__ score, float* __restrict__ smax, int E) {
  int wave = (int)(((long)blockIdx.x * blockDim.x + threadIdx.x) >> 5);
  int lane = threadIdx.x & 31;
  long base = (long)wave * 16;
  if (base >= E) return;                       // wave-uniform; EXEC stays full
  int j = lane & 15;
  long e = base + j;
  if (e >= E) e = E - 1;                       // defensive clamp, keeps EXEC full
  const float* pN = ent + (long)tail[e] * 64;  // A row: neighbor embedding
  const float* pR = rel + (long)(etype[e] - 1) * 64;  // B row: relation embedding
  int half = (lane >> 4) * 2;                  // lanes 0-15: K pair {4k,4k+1}; 16-31: {4k+2,4k+3}
  v8f c = {};
#pragma unroll
  for (int k = 0; k < 16; ++k) {
    int ko = 4 * k + half;
    v2f a = *(const v2f*)(pN + ko);
    v2f b = *(const v2f*)(pR + ko);
    c = __builtin_amdgcn_wmma_f32_16x16x4_f32(false, a, false, b, (short)0, c,
                                              false, false);
  }
  float d = wmma_diag(c, lane) * 0.125f;       // /8.0 per reference
  if (lane < 16 && base + lane < E) {
    score[base + lane] = d;
    unsafeAtomicMax(&smax[head[base + lane]], d);
  }
}

// ---------------------------------------------------------------------------
// Pass 1 (UI): att = dot(user_emb[u] * interaction_emb[t], entity_emb[i]),
// same WMMA diagonal trick with A = U∘I, B = E.
// ---------------------------------------------------------------------------
__global__ __launch_bounds__(256) void ui_score_kernel(
    const float* __restrict__ ent, const float* __restrict__ usr,
    const float* __restrict__ inter,
    const int* __restrict__ uidx, const int* __restrict__ iidx,
    const int* __restrict__ itype,
    float* __restrict__ score, float* __restrict__ smax, int E) {
  int wave = (int)(((long)blockIdx.x * blockDim.x + threadIdx.x) >> 5);
  int lane = threadIdx.x & 31;
  long base = (long)wave * 16;
  if (base >= E) return;
  int j = lane & 15;
  long e = base + j;
  if (e >= E) e = E - 1;
  int u = uidx[e];
  const float* pU = usr + (long)u * 64;
  const float* pI = inter + (long)itype[e] * 64;
  const float* pE = ent + (long)iidx[e] * 64;
  int half = (lane >> 4) * 2;
  v8f c = {};
#pragma unroll
  for (int k = 0; k < 16; ++k) {
    int ko = 4 * k + half;
    v2f uu = *(const v2f*)(pU + ko);
    v2f ii = *(const v2f*)(pI + ko);
    v2f a = uu * ii;                           // elementwise U∘I
    v2f b = *(const v2f*)(pE + ko);
    c = __builtin_amdgcn_wmma_f32_16x16x4_f32(false, a, false, b, (short)0, c,
                                              false, false);
  }
  float d = wmma_diag(c, lane);
  if (lane < 16 && base + lane < E) {
    score[base + lane] = d;
    unsafeAtomicMax(&smax[uidx[base + lane]], d);
  }
}

// ---------------------------------------------------------------------------
// Pass 2: ex = exp(score - segmax); score <- ex; segsum += ex (hw f32 atomic).
// ---------------------------------------------------------------------------
__global__ __launch_bounds__(256) void expsum_kernel(
    float* __restrict__ score, const int* __restrict__ seg,
    const float* __restrict__ smax, float* __restrict__ ssum, int E) {
  long i = (long)blockIdx.x * blockDim.x + threadIdx.x;
  if (i >= E) return;
  int s = seg[i];
  float ex = expf(score[i] - smax[s]);
  score[i] = ex;
  unsafeAtomicAdd(&ssum[s], ex);
}

// ---------------------------------------------------------------------------
// Pass 3: one wave per edge. w = ex / segsum; write normalized weight and
// scatter agg[seg] += w * src_row (2 f32 hw atomics per lane, 64 per edge).
// ---------------------------------------------------------------------------
__global__ __launch_bounds__(256) void scatter_kernel(
    const float* __restrict__ ent, const int* __restrict__ seg,
    const int* __restrict__ rowidx, float* __restrict__ score,
    const float* __restrict__ ssum, float* __restrict__ agg, int E) {
  int wave = (int)(((long)blockIdx.x * blockDim.x + threadIdx.x) >> 5);
  wave = __builtin_amdgcn_readfirstlane(wave);   // provably wave-uniform -> SGPR
  int lane = threadIdx.x & 31;
  if (wave >= E) return;
  int s = seg[wave];
  float w = score[wave] / ssum[s];
  if (lane == 0) score[wave] = w;                // final normalized att / w1
  const float* src = ent + (long)rowidx[wave] * 64 + lane * 2;
  float* dst = agg + (long)s * 64 + lane * 2;
  unsafeAtomicAdd(dst + 0, src[0] * w);
  unsafeAtomicAdd(dst + 1, src[1] * w);
}

// ---------------------------------------------------------------------------
extern "C" void kernel_launch(void* const* d_in, const int* in_sizes, int n_in,
                              void* d_out, int out_size, void* d_ws, size_t ws_size,
                              hipStream_t stream) {
  const float* ent   = (const float*)d_in[0];   // [N_ENT,64]
  const float* usr   = (const float*)d_in[1];   // [N_USR,64]
  const float* inter = (const float*)d_in[2];   // [8,64]
  const float* rel   = (const float*)d_in[3];   // [16,64]
  const int* eidx    = (const int*)d_in[4];     // [2,E_KG]
  const int* etype   = (const int*)d_in[5];     // [E_KG]
  const int* uindex  = (const int*)d_in[6];     // [E_UI]
  const int* iindex  = (const int*)d_in[7];     // [E_UI]
  const int* itype   = (const int*)d_in[8];     // [E_UI]

  const int nEnt = in_sizes[0] / 64;
  const int nUsr = in_sizes[1] / 64;
  const int E_kg = in_sizes[4] / 2;
  const int E_ui = in_sizes[6];
  const int* head = eidx;
  const int* tail = eidx + E_kg;

  // d_out layout (reference return order): entity_agg, user_agg, att, w1
  float* out        = (float*)d_out;
  float* entity_agg = out;
  float* user_agg   = entity_agg + (long)nEnt * 64;
  float* att        = user_agg + (long)nUsr * 64;
  float* w1         = att + E_ui;

  // workspace: per-segment max / sum for both sides (~1.3 MB)
  float* smax_e = (float*)d_ws;
  float* ssum_e = smax_e + nEnt;
  float* smax_u = ssum_e + nEnt;
  float* ssum_u = smax_u + nUsr;

  long nzero = (long)nEnt * 64 + (long)nUsr * 64;
  init_kernel<<<4096, 256, 0, stream>>>(out, nzero, smax_e, ssum_e, nEnt,
                                        smax_u, ssum_u, nUsr);

  {
    long waves = ((long)E_kg + 15) / 16;
    int blocks = (int)((waves * 32 + 255) / 256);
    kg_score_kernel<<<blocks, 256, 0, stream>>>(ent, rel, head, tail, etype,
                                                w1, smax_e, E_kg);
  }
  {
    long waves = ((long)E_ui + 15) / 16;
    int blocks = (int)((waves * 32 + 255) / 256);
    ui_score_kernel<<<blocks, 256, 0, stream>>>(ent, usr, inter, uindex, iindex,
                                                itype, att, smax_u, E_ui);
  }

  expsum_kernel<<<(E_kg + 255) / 256, 256, 0, stream>>>(w1, head, smax_e, ssum_e, E_kg);
  expsum_kernel<<<(E_ui + 255) / 256, 256, 0, stream>>>(att, uindex, smax_u, ssum_u, E_ui);

  {
    int blocks = (int)(((long)E_kg * 32 + 255) / 256);
    scatter_kernel<<<blocks, 256, 0, stream>>>(ent, head, tail, w1, ssum_e,
                                               entity_agg, E_kg);
  }
  {
    int blocks = (int)(((long)E_ui * 32 + 255) / 256);
    scatter_kernel<<<blocks, 256, 0, stream>>>(ent, uindex, iindex, att, ssum_u,
                                               user_agg, E_ui);
  }
}